// TaskAlignedAssigner_11836929867785
// MI455X (gfx1250) — compile-verified
//
#include <hip/hip_runtime.h>
#include <math.h>
#include <stdint.h>

// ---------------- problem constants (from reference) ----------------
constexpr int   B_   = 16;
constexpr int   A_   = 8400;
constexpr int   M_   = 64;
constexpr int   NC_  = 80;
constexpr int   TK_  = 10;     // TOPK
constexpr float EPS_ = 1e-9f;
#define BDIM 256

// ---------------- CDNA5 async global->LDS staging -------------------
// Use inline asm (portable across ROCm7.2 / amdgpu-toolchain per bridge doc).
// VDST = VGPR holding per-lane LDS byte address (low 32 bits of the generic
// shared pointer == wave-relative LDS address per the aperture mapping).
#if defined(__HIP_DEVICE_COMPILE__) && defined(__gfx1250__)
#define ATA_ASYNC 1
#else
#define ATA_ASYNC 0
#endif

__device__ __forceinline__ void stage_b32(const unsigned* g, unsigned* l) {
#if ATA_ASYNC
  unsigned           lds_addr = (unsigned)(unsigned long long)(uintptr_t)l;
  unsigned long long gaddr    = (unsigned long long)(uintptr_t)g;
  asm volatile("global_load_async_to_lds_b32 %0, %1, off"
               :: "v"(lds_addr), "v"(gaddr) : "memory");
#else
  *l = *g;
#endif
}
__device__ __forceinline__ void stage_wait() {
#if ATA_ASYNC
  asm volatile("s_wait_asynccnt 0" ::: "memory");
#endif
}

// ---------------- math helpers --------------------------------------
__device__ __forceinline__ float in_gt_min(float ax, float ay,
                                           float x1, float y1, float x2, float y2) {
  return fminf(fminf(ax - x1, ay - y1), fminf(x2 - ax, y2 - ay));
}

// CIoU(box1=gt, box2=pred) clipped to [0, inf), exactly mirroring reference _ciou.
__device__ __forceinline__ float ciou_clip(float gx1, float gy1, float gx2, float gy2,
                                           float px1, float py1, float px2, float py2) {
  const float e  = 1e-7f;
  float w1 = gx2 - gx1, h1 = gy2 - gy1 + e;
  float w2 = px2 - px1, h2 = py2 - py1 + e;
  float iw = fmaxf(fminf(gx2, px2) - fmaxf(gx1, px1), 0.f);
  float ih = fmaxf(fminf(gy2, py2) - fmaxf(gy1, py1), 0.f);
  float inter = iw * ih;
  float uni   = w1 * h1 + w2 * h2 - inter + e;
  float iou   = inter / uni;
  float cw = fmaxf(gx2, px2) - fminf(gx1, px1);
  float ch = fmaxf(gy2, py2) - fminf(gy1, py1);
  float c2 = cw * cw + ch * ch + e;
  float dx = px1 + px2 - gx1 - gx2;
  float dy = py1 + py2 - gy1 - gy2;
  float rho2 = (dx * dx + dy * dy) * 0.25f;
  float dat  = atanf(w2 / h2) - atanf(w1 / h1);
  float v    = 0.4052847345693511f * dat * dat;   // 4/pi^2
  float alpha = v / (v - iou + 1.0f + e);
  float ciou  = iou - (rho2 / c2 + v * alpha);
  return fmaxf(ciou, 0.f);
}

__device__ __forceinline__ float ov6f(float ov) { float o2 = ov * ov; return o2 * o2 * o2; }

__device__ __forceinline__ void atomicMaxNonNegF(float* addr, float v) {
  // values are >= 0 -> IEEE order == unsigned bit order
  atomicMax((unsigned*)addr, __float_as_uint(v));
}

// ---------------- kernel 1: per-(b,m) top-k --------------------------
__global__ __launch_bounds__(BDIM)
void topk_kernel(const float* __restrict__ pd_scores, const float* __restrict__ pd_bboxes,
                 const float* __restrict__ anc, const int* __restrict__ gt_labels,
                 const float* __restrict__ gt_bboxes, const float* __restrict__ mask_gt,
                 int* __restrict__ lists) {
  __shared__ float s_align[A_];
  __shared__ float rv[BDIM];
  __shared__ int   ri[BDIM];
  __shared__ int   s_out[TK_ + 1];

  const int bm  = blockIdx.x;           // b*M + m
  const int b   = bm / M_;
  const int tid = threadIdx.x;

  const float mg = mask_gt[bm];
  if (mg <= 0.f) {                       // padded gt: row is all-zero -> nothing kept
    if (tid < TK_ + 1) lists[bm * (TK_ + 1) + tid] = -1;
    return;
  }
  const int   lab = gt_labels[bm];
  const float gx1 = gt_bboxes[bm * 4 + 0];
  const float gy1 = gt_bboxes[bm * 4 + 1];
  const float gx2 = gt_bboxes[bm * 4 + 2];
  const float gy2 = gt_bboxes[bm * 4 + 3];

  // compute masked align row into LDS
  for (int a = tid; a < A_; a += BDIM) {
    float ax = anc[2 * a + 0], ay = anc[2 * a + 1];
    float al = 0.f;
    if (in_gt_min(ax, ay, gx1, gy1, gx2, gy2) > EPS_) {
      const float* pb = pd_bboxes + ((size_t)b * A_ + a) * 4;
      float ov = ciou_clip(gx1, gy1, gx2, gy2, pb[0], pb[1], pb[2], pb[3]);
      float sc = pd_scores[((size_t)b * A_ + a) * NC_ + lab];
      al = sqrtf(sc) * ov6f(ov);
    }
    s_align[a] = al;
  }
  __syncthreads();

  // 10 sequential block-wide argmax passes (lowest-index tie break)
  for (int k = 0; k < TK_; ++k) {
    float bv = -1.f; int bi = 0;
    for (int a = tid; a < A_; a += BDIM) {
      float v = s_align[a];
      if (v > bv) { bv = v; bi = a; }
    }
    rv[tid] = bv; ri[tid] = bi;
    __syncthreads();
    for (int s = BDIM / 2; s > 0; s >>= 1) {
      if (tid < s) {
        float v = rv[tid + s]; int i2 = ri[tid + s];
        if (v > rv[tid] || (v == rv[tid] && i2 < ri[tid])) { rv[tid] = v; ri[tid] = i2; }
      }
      __syncthreads();
    }
    if (tid == 0) {
      int wi = ri[0];
      s_out[k] = (rv[0] > EPS_) ? wi : -1;   // keep <=> metric > EPS (masked rows are 0)
      s_align[wi] = -1.f;                    // exclude from later passes
    }
    __syncthreads();
  }

  if (tid == 0) {
    // reference quirk: dropped slots scatter-add at anchor 0; survives only if
    // mask_pos_gt[b,m,0] (mg>0 here, so just the in-gt test for anchor 0).
    bool dropped = false;
    for (int k = 0; k < TK_; ++k) dropped |= (s_out[k] < 0);
    int extra = -1;
    if (dropped) {
      float ax = anc[0], ay = anc[1];
      if (in_gt_min(ax, ay, gx1, gy1, gx2, gy2) > EPS_) extra = 0;
    }
    s_out[TK_] = extra;
    for (int k = 0; k <= TK_; ++k) lists[bm * (TK_ + 1) + k] = s_out[k];
  }
}

// ---------------- kernel 2: per-anchor assignment --------------------
__global__ __launch_bounds__(BDIM)
void assign_kernel(const float* __restrict__ pd_scores, const float* __restrict__ pd_bboxes,
                   const float* __restrict__ anc, const int* __restrict__ gt_labels,
                   const float* __restrict__ gt_bboxes, const float* __restrict__ mask_gt,
                   const int* __restrict__ lists,
                   int* __restrict__ tgt, int* __restrict__ fgbuf, float* __restrict__ align_at,
                   float* __restrict__ pos_al, float* __restrict__ pos_ov,
                   float* __restrict__ out_bboxes, float* __restrict__ out_fg) {
  __shared__ float s_gtb[M_ * 4];
  __shared__ int   s_lab[M_];
  __shared__ float s_mgt[M_];
  __shared__ int   s_list[M_ * (TK_ + 1)];

  const int b   = blockIdx.y;
  const int tid = threadIdx.x;
  const int a   = blockIdx.x * BDIM + tid;

  // ---- async-stage per-batch gt data + topk lists into LDS ----
  {
    const unsigned* gb = (const unsigned*)(gt_bboxes + (size_t)b * M_ * 4);
    for (int i = tid; i < M_ * 4; i += BDIM) stage_b32(gb + i, (unsigned*)s_gtb + i);
    const unsigned* gl = (const unsigned*)(gt_labels + (size_t)b * M_);
    if (tid < M_) stage_b32(gl + tid, (unsigned*)s_lab + tid);
    const unsigned* gm = (const unsigned*)(mask_gt + (size_t)b * M_);
    if (tid < M_) stage_b32(gm + tid, (unsigned*)s_mgt + tid);
    const unsigned* ll = (const unsigned*)(lists + (size_t)b * M_ * (TK_ + 1));
    for (int i = tid; i < M_ * (TK_ + 1); i += BDIM) stage_b32(ll + i, (unsigned*)s_list + i);
    stage_wait();
    __syncthreads();
  }
  if (a >= A_) return;

  // membership of this anchor in each gt's kept set
  int cnt = 0, first = -1;
  for (int m = 0; m < M_; ++m) {
    const int* row = s_list + m * (TK_ + 1);
    bool mem = false;
    #pragma unroll
    for (int s = 0; s < TK_ + 1; ++s) mem |= (row[s] == a);
    if (mem) { ++cnt; if (first < 0) first = m; }
  }

  const float  ax = anc[2 * a + 0], ay = anc[2 * a + 1];
  const float* pb = pd_bboxes + ((size_t)b * A_ + a) * 4;
  const float  px1 = pb[0], py1 = pb[1], px2 = pb[2], py2 = pb[3];

  int t = 0;
  if (cnt == 1) {
    t = first;
  } else if (cnt > 1) {
    // multi-assigned: argmax over masked overlaps across ALL m (reference is_max)
    float bv = -1.f; int bi = 0;
    for (int m = 0; m < M_; ++m) {
      float ov = 0.f;
      if (s_mgt[m] > 0.f) {
        float x1 = s_gtb[m * 4 + 0], y1 = s_gtb[m * 4 + 1];
        float x2 = s_gtb[m * 4 + 2], y2 = s_gtb[m * 4 + 3];
        if (in_gt_min(ax, ay, x1, y1, x2, y2) > EPS_)
          ov = ciou_clip(x1, y1, x2, y2, px1, py1, px2, py2);
      }
      if (ov > bv) { bv = ov; bi = m; }   // strict > -> lowest index on ties
    }
    t = bi;
  }

  const int fg = (cnt > 0) ? 1 : 0;
  float alv = 0.f;
  if (fg) {
    float ov = 0.f;
    float x1 = s_gtb[t * 4 + 0], y1 = s_gtb[t * 4 + 1];
    float x2 = s_gtb[t * 4 + 2], y2 = s_gtb[t * 4 + 3];
    if (s_mgt[t] > 0.f && in_gt_min(ax, ay, x1, y1, x2, y2) > EPS_)
      ov = ciou_clip(x1, y1, x2, y2, px1, py1, px2, py2);
    float sc = pd_scores[((size_t)b * A_ + a) * NC_ + s_lab[t]];
    alv = sqrtf(sc) * ov6f(ov);           // masked automatically (ov==0 when mask false)
    atomicMaxNonNegF(&pos_al[b * M_ + t], alv);
    atomicMaxNonNegF(&pos_ov[b * M_ + t], ov);
  }

  const size_t ia = (size_t)b * A_ + a;
  tgt[ia]      = t;
  fgbuf[ia]    = fg;
  align_at[ia] = alv;
  float4 bb = make_float4(s_gtb[t * 4 + 0], s_gtb[t * 4 + 1],
                          s_gtb[t * 4 + 2], s_gtb[t * 4 + 3]);
  *(float4*)(out_bboxes + ia * 4) = bb;   // gathered even when fg==0 (reference)
  out_fg[ia] = fg ? 1.f : 0.f;
}

// ---------------- kernel 3: normalized one-hot scores ----------------
__global__ __launch_bounds__(BDIM)
void scores_kernel(const int* __restrict__ gt_labels, const int* __restrict__ tgt,
                   const int* __restrict__ fgbuf, const float* __restrict__ align_at,
                   const float* __restrict__ pos_al, const float* __restrict__ pos_ov,
                   float* __restrict__ out_scores) {
  const int i = blockIdx.x * BDIM + threadIdx.x;   // b*A + a
  if (i >= B_ * A_) return;
  const int b = i / A_;
  float norm = 0.f; int lab = -1;
  if (fgbuf[i]) {
    int t = tgt[i];
    lab  = gt_labels[b * M_ + t];
    norm = align_at[i] * pos_ov[b * M_ + t] / (pos_al[b * M_ + t] + EPS_);
  }
  float* o = out_scores + (size_t)i * NC_;
  #pragma unroll
  for (int c = 0; c < NC_; c += 4) {
    float4 v;
    v.x = (c + 0 == lab) ? norm : 0.f;
    v.y = (c + 1 == lab) ? norm : 0.f;
    v.z = (c + 2 == lab) ? norm : 0.f;
    v.w = (c + 3 == lab) ? norm : 0.f;
    *(float4*)(o + c) = v;
  }
}

// ---------------- host launcher --------------------------------------
extern "C" void kernel_launch(void* const* d_in, const int* in_sizes, int n_in,
                              void* d_out, int out_size, void* d_ws, size_t ws_size,
                              hipStream_t stream) {
  (void)in_sizes; (void)n_in; (void)out_size; (void)ws_size;
  const float* pd_scores = (const float*)d_in[0];
  const float* pd_bboxes = (const float*)d_in[1];
  const float* anc       = (const float*)d_in[2];
  const int*   gt_labels = (const int*)  d_in[3];
  const float* gt_bboxes = (const float*)d_in[4];
  const float* mask_gt   = (const float*)d_in[5];

  float* out        = (float*)d_out;
  float* out_bboxes = out;                                  // B*A*4
  float* out_scores = out + (size_t)B_ * A_ * 4;            // B*A*NC
  float* out_fg     = out_scores + (size_t)B_ * A_ * NC_;   // B*A

  char* p = (char*)d_ws;
  auto take = [&](size_t bytes) -> void* {
    void* r = (void*)p; p += (bytes + 15) & ~(size_t)15; return r;
  };
  int*   lists    = (int*)  take(sizeof(int)   * B_ * M_ * (TK_ + 1));
  int*   tgt      = (int*)  take(sizeof(int)   * (size_t)B_ * A_);
  int*   fgbuf    = (int*)  take(sizeof(int)   * (size_t)B_ * A_);
  float* align_at = (float*)take(sizeof(float) * (size_t)B_ * A_);
  float* pos_al   = (float*)take(sizeof(float) * B_ * M_);
  float* pos_ov   = (float*)take(sizeof(float) * B_ * M_);

  topk_kernel<<<dim3(B_ * M_), dim3(BDIM), 0, stream>>>(
      pd_scores, pd_bboxes, anc, gt_labels, gt_bboxes, mask_gt, lists);

  hipMemsetAsync(pos_al, 0, sizeof(float) * B_ * M_, stream);
  hipMemsetAsync(pos_ov, 0, sizeof(float) * B_ * M_, stream);

  assign_kernel<<<dim3((A_ + BDIM - 1) / BDIM, B_), dim3(BDIM), 0, stream>>>(
      pd_scores, pd_bboxes, anc, gt_labels, gt_bboxes, mask_gt, lists,
      tgt, fgbuf, align_at, pos_al, pos_ov, out_bboxes, out_fg);

  scores_kernel<<<dim3((B_ * A_ + BDIM - 1) / BDIM), dim3(BDIM), 0, stream>>>(
      gt_labels, tgt, fgbuf, align_at, pos_al, pos_ov, out_scores);
}